// Conv2d_62405874811871
// MI455X (gfx1250) — compile-verified
//
#include <hip/hip_runtime.h>
#include <hip/hip_bf16.h>

// CDNA5 / gfx1250, wave32. D = A(16x4 f32) x B(4x16 f32) + C(16x16 f32).
typedef __attribute__((ext_vector_type(2))) float v2f;
typedef __attribute__((ext_vector_type(4))) float v4f;
typedef __attribute__((ext_vector_type(8))) float v8f;

#define OC      512
#define KSZ     49
#define CTILES  (OC / 16)   // 32 channel tiles of 16
#define KFULL   12          // full K=4 steps covering k=0..47
// step 12 handles k=48 (k=49..51 zero-padded)

__global__ __launch_bounds__(256) void conv_im2col_wmma_f32(
    const float* __restrict__ x,     // [windows, 49]  im2col matrix
    const float* __restrict__ wgt,   // [512, 49]      flattened 7x7 kernels
    const float* __restrict__ bias,  // [512]
    float* __restrict__ out,         // [512, windows] packed
    int windows)
{
    const int lane = threadIdx.x & 31;
    const int wave = threadIdx.x >> 5;          // 0..7
    const int lo   = lane & 15;                 // row/col within half-wave
    const int hi   = lane >> 4;                 // 0: K={0,1}, 1: K={2,3}

    // Each wave owns 32 consecutive windows = two 16-wide N tiles.
    const long wBase = (long)blockIdx.x * 256 + (long)wave * 32;

    // ---- Preload B fragments (enc_x^T, K x 16 windows) into registers ----
    // B frag VGPR0: lanes0-15 K=4s+0, lanes16-31 K=4s+2 ; VGPR1: K=4s+1 / 4s+3.
    // Per lane that is a contiguous float2 at row[4s + 2*hi].
    v2f B[2][KFULL + 1];
    #pragma unroll
    for (int t = 0; t < 2; ++t) {
        const float* row = x + (size_t)(wBase + t * 16 + lo) * KSZ;
        #pragma unroll
        for (int s = 0; s < KFULL; ++s) {
            B[t][s] = __builtin_nontemporal_load((const v2f*)(row + 4 * s + 2 * hi));
        }
        // Tail step: k=48 valid for every row; k=49..51 are zero.
        float k48 = __builtin_nontemporal_load(row + 48);
        v2f bt; bt.x = (hi == 0) ? k48 : 0.0f; bt.y = 0.0f;
        B[t][KFULL] = bt;
    }

    // ---- Loop over all 32 channel tiles, reusing the register-resident B ----
    for (int c = 0; c < CTILES; ++c) {
        const float* wrow = wgt + (size_t)(c * 16 + lo) * KSZ;  // A row = channel
        v8f acc0 = {};
        v8f acc1 = {};

        #pragma unroll
        for (int s = 0; s < KFULL; ++s) {
            v2f A = *(const v2f*)(wrow + 4 * s + 2 * hi);  // weights: L2-hot
            acc0 = __builtin_amdgcn_wmma_f32_16x16x4_f32(
                       false, A, false, B[0][s], (short)0, acc0, false, false);
            acc1 = __builtin_amdgcn_wmma_f32_16x16x4_f32(
                       false, A, false, B[1][s], (short)0, acc1, false, false);
        }
        {   // Tail step k=48 (zero-padded to K=52 on both operands)
            float k48 = wrow[48];
            v2f A; A.x = (hi == 0) ? k48 : 0.0f; A.y = 0.0f;
            acc0 = __builtin_amdgcn_wmma_f32_16x16x4_f32(
                       false, A, false, B[0][KFULL], (short)0, acc0, false, false);
            acc1 = __builtin_amdgcn_wmma_f32_16x16x4_f32(
                       false, A, false, B[1][KFULL], (short)0, acc1, false, false);
        }

        // ---- Bias + store ----
        // D VGPR r holds channel (c*16 + hi*8 + r), window (wBase + t*16 + lo).
        const float* bp = bias + c * 16 + hi * 8;    // 16B-aligned, broadcast load
        v4f b0 = *(const v4f*)(bp);
        v4f b1 = *(const v4f*)(bp + 4);
        float bv[8] = { b0.x, b0.y, b0.z, b0.w, b1.x, b1.y, b1.z, b1.w };

        const long col = wBase + lo;
        #pragma unroll
        for (int r = 0; r < 8; ++r) {
            const size_t chan = (size_t)(c * 16 + hi * 8 + r);
            float* o = out + chan * (size_t)windows + col;
            __builtin_nontemporal_store(acc0[r] + bv[r], o);        // windows [wBase, +16)
            __builtin_nontemporal_store(acc1[r] + bv[r], o + 16);   // windows [wBase+16, +32)
        }
    }
}

extern "C" void kernel_launch(void* const* d_in, const int* in_sizes, int n_in,
                              void* d_out, int out_size, void* d_ws, size_t ws_size,
                              hipStream_t stream) {
    const float* x    = (const float*)d_in[0];   // enc_x  [windows, 49] f32
    const float* wgt  = (const float*)d_in[1];   // weight [512, 7, 7]   f32
    const float* bias = (const float*)d_in[2];   // bias   [512]         f32
    float* out        = (float*)d_out;           // [512 * windows]      f32

    const int windows = in_sizes[0] / KSZ;       // 262144
    const int blocks  = windows / 256;           // 1024 (windows is a multiple of 256)

    conv_im2col_wmma_f32<<<blocks, 256, 0, stream>>>(x, wgt, bias, out, windows);
}